// SEBlock_27084063768970
// MI455X (gfx1250) — compile-verified
//
#include <hip/hip_runtime.h>
#include <math.h>

// SE block: B=64, L=4096, C=256, H=32
// ws layout: [0, 64*256) floats = Ssum (column sums), [64*256, 2*64*256) = G (gates)

typedef __attribute__((ext_vector_type(2))) float v2f;
typedef __attribute__((ext_vector_type(8))) float v8f;

#define SE_B 64
#define SE_L 4096
#define SE_C 256
#define SE_H 32

// ---------------- zero the accumulator buffer ----------------
__global__ __launch_bounds__(256) void se_zero(float* __restrict__ Ssum) {
    Ssum[blockIdx.x * 256 + threadIdx.x] = 0.0f;
}

// ---------------- pass 1: column sums over L ----------------
// grid = B * CHUNKS, block = 256 (one thread per channel)
__global__ __launch_bounds__(256) void se_reduce(const float* __restrict__ in,
                                                 float* __restrict__ Ssum) {
    const int CHUNKS = 64;
    const int ROWS = SE_L / CHUNKS;  // 64 rows per block
    int b = blockIdx.x / CHUNKS;
    int chunk = blockIdx.x % CHUNKS;
    int c = threadIdx.x;
    const float* p = in + ((size_t)b * SE_L + (size_t)chunk * ROWS) * SE_C + c;
    float s = 0.0f;
#pragma unroll 8
    for (int i = 0; i < ROWS; ++i) s += p[(size_t)i * SE_C];
    atomicAdd(&Ssum[b * SE_C + c], s);
}

// ---------------- pass 2: gate via fp32 WMMA ----------------
// single block, 256 threads = 8 waves (wave32)
__global__ __launch_bounds__(256) void se_gate(const float* __restrict__ Ssum,
                                               const float* __restrict__ w1,
                                               const float* __restrict__ b1,
                                               const float* __restrict__ w2,
                                               const float* __restrict__ b2,
                                               float* __restrict__ G) {
    __shared__ float E[SE_B * SE_H];  // 64 x 32 fp32 = 8 KB

    const int wave = threadIdx.x >> 5;   // 0..7 (uniform per wave)
    const int lane = threadIdx.x & 31;
    const int half = lane >> 4;          // 0 or 1
    const int m16 = lane & 15;
    const float invL = 1.0f / (float)SE_L;

    // ---- Phase A: E = relu(mean(S) @ w1 + b1);  M=64,N=32,K=256 ----
    // 8 tiles of 16x16, one per wave: mt = wave>>1 (0..3), nt = wave&1 (0..1)
    {
        const int mt = wave >> 1;
        const int nt = wave & 1;
        const int m = mt * 16 + m16;
        const int n = nt * 16 + m16;
        v8f acc = {};
#pragma unroll 4
        for (int kk = 0; kk < SE_C; kk += 4) {
            const int k = kk + 2 * half;  // lanes 0-15: K=kk,kk+1 ; lanes 16-31: K=kk+2,kk+3
            v2f a, bf;
            a.x = Ssum[m * SE_C + k] * invL;
            a.y = Ssum[m * SE_C + k + 1] * invL;
            bf.x = w1[k * SE_H + n];
            bf.y = w1[(k + 1) * SE_H + n];
            acc = __builtin_amdgcn_wmma_f32_16x16x4_f32(
                false, a, false, bf, (short)0, acc, false, false);
        }
        const float bias = b1[n];
#pragma unroll
        for (int r = 0; r < 8; ++r) {
            const int mrow = mt * 16 + r + 8 * half;  // C/D layout: M = r + 8*(lane>=16)
            float v = acc[r] + bias;
            E[mrow * SE_H + n] = v > 0.0f ? v : 0.0f;
        }
    }
    __syncthreads();

    // ---- Phase B: G = sigmoid(E @ w2 + b2);  M=64,N=256,K=32 ----
    // 64 tiles of 16x16; each wave does 8
    for (int t = wave; t < 64; t += 8) {
        const int mt = t >> 4;      // 0..3
        const int ntile = t & 15;   // 0..15
        const int m = mt * 16 + m16;
        const int n = ntile * 16 + m16;
        v8f acc = {};
#pragma unroll
        for (int kk = 0; kk < SE_H; kk += 4) {
            const int k = kk + 2 * half;
            v2f a, bf;
            a.x = E[m * SE_H + k];
            a.y = E[m * SE_H + k + 1];
            bf.x = w2[k * SE_C + n];
            bf.y = w2[(k + 1) * SE_C + n];
            acc = __builtin_amdgcn_wmma_f32_16x16x4_f32(
                false, a, false, bf, (short)0, acc, false, false);
        }
        const float bias = b2[n];
#pragma unroll
        for (int r = 0; r < 8; ++r) {
            const int mrow = mt * 16 + r + 8 * half;
            float v = acc[r] + bias;
            G[mrow * SE_C + n] = 1.0f / (1.0f + __expf(-v));
        }
    }
}

// ---------------- pass 3: scale ----------------
// one float4 per thread; 2^24 float4s -> 65536 blocks of 256
__global__ __launch_bounds__(256) void se_scale(const float* __restrict__ in,
                                                const float* __restrict__ G,
                                                float* __restrict__ out) {
    const size_t idx = (size_t)blockIdx.x * 256 + threadIdx.x;
    const size_t e = idx * 4;                 // element base
    const int c = (int)(e & (SE_C - 1));      // channel (multiple of 4)
    const int b = (int)(e >> 20);             // e / (L*C) = e / 2^20
    const float4 v = *(const float4*)(in + e);
    const float4 g = *(const float4*)(G + b * SE_C + c);
    float4 o;
    o.x = v.x * g.x;
    o.y = v.y * g.y;
    o.z = v.z * g.z;
    o.w = v.w * g.w;
    *(float4*)(out + e) = o;
}

extern "C" void kernel_launch(void* const* d_in, const int* in_sizes, int n_in,
                              void* d_out, int out_size, void* d_ws, size_t ws_size,
                              hipStream_t stream) {
    const float* in = (const float*)d_in[0];
    const float* w1 = (const float*)d_in[1];
    const float* b1 = (const float*)d_in[2];
    const float* w2 = (const float*)d_in[3];
    const float* b2 = (const float*)d_in[4];
    float* out = (float*)d_out;

    float* Ssum = (float*)d_ws;                 // 64*256 floats
    float* G = Ssum + SE_B * SE_C;              // 64*256 floats

    se_zero<<<SE_B, 256, 0, stream>>>(Ssum);
    se_reduce<<<SE_B * 64, 256, 0, stream>>>(in, Ssum);
    se_gate<<<1, 256, 0, stream>>>(Ssum, w1, b1, w2, b2, G);
    se_scale<<<(SE_B * SE_L * SE_C) / 4 / 256, 256, 0, stream>>>(in, G, out);
}